// Group_28063316312782
// MI455X (gfx1250) — compile-verified
//
#include <hip/hip_runtime.h>
#include <hip/hip_bf16.h>

// Problem constants (from reference)
#define BATCH      32
#define NPTS       8192
#define NGROUP     512
#define GSIZE      32        // k of kNN == wave32 lane count
#define NC         256       // points per KNN chunk
#define PADROW     (NC + 4)  // padded LDS row stride (bank-conflict avoidance)

typedef __attribute__((ext_vector_type(2))) float v2f;
typedef __attribute__((ext_vector_type(8))) float v8f;

// ---------------------------------------------------------------------------
// Wave32 cross-lane XOR exchange — pure VALU (no LDS pipe, no DScnt waits):
//   mask 1/2/4 : v_mov_b32 dpp8 (permute within 8 lanes)
//   mask 8     : v_permlane16_b32 (lane^8 inside each 16-lane row)
//   mask 16    : v_permlanex16_b32 identity (swap 16-lane halves)
//   mask 31    : v_permlanex16_b32 mirror (full 32-lane reverse)
// ---------------------------------------------------------------------------
#define DPP8SEL(a,b,c,d,e,f,g,h) \
    ((a)|((b)<<3)|((c)<<6)|((d)<<9)|((e)<<12)|((f)<<15)|((g)<<18)|((h)<<21))

template <int MASK> __device__ __forceinline__ int lanex_i(int v);
template <> __device__ __forceinline__ int lanex_i<1>(int v) {
    return __builtin_amdgcn_mov_dpp8(v, DPP8SEL(1,0,3,2,5,4,7,6));
}
template <> __device__ __forceinline__ int lanex_i<2>(int v) {
    return __builtin_amdgcn_mov_dpp8(v, DPP8SEL(2,3,0,1,6,7,4,5));
}
template <> __device__ __forceinline__ int lanex_i<4>(int v) {
    return __builtin_amdgcn_mov_dpp8(v, DPP8SEL(4,5,6,7,0,1,2,3));
}
template <> __device__ __forceinline__ int lanex_i<8>(int v) {
    // lanes 0-7 <- 8-15, lanes 8-15 <- 0-7 (within each 16-lane row)
    return __builtin_amdgcn_permlane16(v, v, 0xFEDCBA98, 0x76543210, false, false);
}
template <> __device__ __forceinline__ int lanex_i<16>(int v) {
    // identity selector across the opposite 16-lane half
    return __builtin_amdgcn_permlanex16(v, v, 0x76543210, 0xFEDCBA98, false, false);
}
template <> __device__ __forceinline__ int lanex_i<31>(int v) {
    // mirror selector across the opposite half: lane i <- lane 31-i
    return __builtin_amdgcn_permlanex16(v, v, 0x89ABCDEF, 0x01234567, false, false);
}
template <int MASK> __device__ __forceinline__ float lanex_f(float v) {
    return __int_as_float(lanex_i<MASK>(__float_as_int(v)));
}

// Branchless compare-exchange: swap iff signflip(pd) < signflip(d).
// sgn == 0          -> keep-min lane;  sgn == 0x80000000 -> keep-max lane.
// Equality never swaps on either side, so (dist,idx) pairs stay intact.
template <int MASK>
__device__ __forceinline__ void cmpSwap(float& d, int& i, int sgn) {
    const float pd = lanex_f<MASK>(d);
    const int   pi = lanex_i<MASK>(i);
    const bool  sw = __int_as_float(__float_as_int(pd) ^ sgn) <
                     __int_as_float(__float_as_int(d)  ^ sgn);
    d = sw ? pd : d;
    i = sw ? pi : i;
}

__device__ __forceinline__ int dirSgn(bool keepMin) {
    return keepMin ? 0 : (int)0x80000000u;
}

// Full bitonic sort of 32 (dist,idx) pairs, ascending across lanes.
__device__ __forceinline__ void bitonicSort32Asc(float& d, int& i, int lane) {
#pragma unroll
    for (int k = 2; k <= 32; k <<= 1) {
        const bool asc = ((lane & k) == 0);
#pragma unroll
        for (int j = k >> 1; j > 0; j >>= 1) {
            const int sgn = dirSgn(((lane & j) == 0) == asc);
            switch (j) {
                case 16: cmpSwap<16>(d, i, sgn); break;
                case 8:  cmpSwap<8>(d, i, sgn);  break;
                case 4:  cmpSwap<4>(d, i, sgn);  break;
                case 2:  cmpSwap<2>(d, i, sgn);  break;
                default: cmpSwap<1>(d, i, sgn);  break;
            }
        }
    }
}

// Merge a bitonic 32-sequence into ascending order.
__device__ __forceinline__ void bitonicMerge32Asc(float& d, int& i, int lane) {
    cmpSwap<16>(d, i, dirSgn((lane & 16) == 0));
    cmpSwap<8>(d, i, dirSgn((lane & 8) == 0));
    cmpSwap<4>(d, i, dirSgn((lane & 4) == 0));
    cmpSwap<2>(d, i, dirSgn((lane & 2) == 0));
    cmpSwap<1>(d, i, dirSgn((lane & 1) == 0));
}

// Butterfly argmax combine (value, lowest-index tie-break, jnp.argmax semantics)
template <int MASK>
__device__ __forceinline__ void argmaxStep(float& v, int& j) {
    const float ov = lanex_f<MASK>(v);
    const int   oj = lanex_i<MASK>(j);
    const bool take = (ov > v) || ((ov == v) && (oj < j));
    v = take ? ov : v;
    j = take ? oj : j;
}
__device__ __forceinline__ void waveArgmax(float& v, int& j) {
    argmaxStep<16>(v, j);
    argmaxStep<8>(v, j);
    argmaxStep<4>(v, j);
    argmaxStep<2>(v, j);
    argmaxStep<1>(v, j);
}

// ---------------------------------------------------------------------------
// Kernel 1: Farthest Point Sampling. One block per batch, 1024 threads
// (32 wave32). Each thread owns 8 points in registers. Sequential 512-step
// scan with block-wide argmax. Emits center coords into d_out center region.
// ---------------------------------------------------------------------------
__global__ __launch_bounds__(1024)
void fps_kernel(const float* __restrict__ xyz, float* __restrict__ out_center) {
    const int b    = blockIdx.x;
    const int tid  = threadIdx.x;
    const int lane = tid & 31;
    const int wv   = tid >> 5;

    __shared__ int   farSh;
    __shared__ float redV[32];
    __shared__ int   redJ[32];

    const size_t bbase = (size_t)b * NPTS * 3;

    float px[8], py[8], pz[8], dd[8];
#pragma unroll
    for (int k = 0; k < 8; ++k) {
        const int j = tid + k * 1024;
        const float* p = xyz + bbase + (size_t)j * 3;
        px[k] = p[0]; py[k] = p[1]; pz[k] = p[2];
        dd[k] = 1e10f;   // INIT_DIST
    }

    if (tid == 0) farSh = 0;
    __syncthreads();

    for (int it = 0; it < NGROUP; ++it) {
        const int far = farSh;

        // emit center = xyz[far] (reference emits the *old* farthest)
        if (tid < 3)
            out_center[((size_t)b * NGROUP + it) * 3 + tid] =
                xyz[bbase + (size_t)far * 3 + tid];

        // uniform (scalar) loads of the centroid coords
        const float cx = xyz[bbase + (size_t)far * 3 + 0];
        const float cy = xyz[bbase + (size_t)far * 3 + 1];
        const float cz = xyz[bbase + (size_t)far * 3 + 2];

        // distance update + thread-local argmax (earliest index on ties)
        float bv = -1.0f; int bj = 0;
#pragma unroll
        for (int k = 0; k < 8; ++k) {
            const float dx = px[k] - cx, dy = py[k] - cy, dz = pz[k] - cz;
            const float d2 = dx * dx + dy * dy + dz * dz;
            dd[k] = fminf(dd[k], d2);
            const int j = tid + k * 1024;
            if (dd[k] > bv) { bv = dd[k]; bj = j; }
        }

        waveArgmax(bv, bj);
        if (lane == 0) { redV[wv] = bv; redJ[wv] = bj; }
        __syncthreads();

        if (tid < 32) {
            float v = redV[tid]; int j = redJ[tid];
            waveArgmax(v, j);
            if (tid == 0) farSh = j;
        }
        __syncthreads();
    }
}

// ---------------------------------------------------------------------------
// Kernel 2: KNN + gather + center-subtract.
// Grid: BATCH * (NGROUP/16) blocks; block: 512 threads = 16 wave32.
// Per chunk of 256 points each wave computes one 16x16 sqdist tile with
// V_WMMA_F32_16X16X4_F32 (A = -2*centers padded to K=4, B = points,
// C preloaded with q2[M]+r2[N]), scatters it to LDS, then each wave streams
// its own center's row maintaining a sorted top-32 (one pair per lane).
// ---------------------------------------------------------------------------
__global__ __launch_bounds__(512)
void knn_group_kernel(const float* __restrict__ xyz,
                      const float* __restrict__ centers,   // (B, NGROUP, 3)
                      float* __restrict__ out_neigh) {     // (B, NGROUP, 32, 3)
    const int b    = blockIdx.x >> 5;          // / (NGROUP/16)
    const int g0   = (blockIdx.x & 31) * 16;   // first center of this tile
    const int tid  = threadIdx.x;
    const int lane = tid & 31;
    const int wv   = tid >> 5;                 // 0..15: tile col-group AND center id
    const int m    = lane & 15;
    const bool hi  = lane >= 16;
    const int rowOff = hi ? 8 : 0;

    __shared__ float sX[NC], sY[NC], sZ[NC], sR2[NC];
    __shared__ float4 sCen[16];                // x,y,z,q2
    __shared__ float sDist[16 * PADROW];

    const size_t bbase = (size_t)b * NPTS * 3;

    if (tid < 16) {
        const float* c = centers + ((size_t)b * NGROUP + g0 + tid) * 3;
        const float x = c[0], y = c[1], z = c[2];
        sCen[tid] = make_float4(x, y, z, x * x + y * y + z * z);
    }
    __syncthreads();

    // A (16x4 f32): lanes 0-15 hold K=0,1 of row M=lane; lanes 16-31 hold
    // K=2,3 of row M=lane-16. Fold -2 into A. Branchless half-wave select.
    const float4 cm = sCen[m];
    v2f av;
    av.x = -2.0f * (hi ? cm.z : cm.x);
    av.y = hi ? 0.0f : (-2.0f * cm.y);

    // q2 per accumulator row: C vgpr r holds M = r (+8 for lanes 16-31)
    float q2v[8];
#pragma unroll
    for (int r = 0; r < 8; ++r) q2v[r] = sCen[rowOff + r].w;

    // running top-32 for center (g0+wv): sorted ascending across lanes
    float bd = 1e30f;
    int   bi = 0;

    for (int cb = 0; cb < NPTS / NC; ++cb) {
        const int chunk = cb * NC;

        // stage chunk of points into LDS (SoA + r2)
        if (tid < NC) {
            const float* p = xyz + bbase + (size_t)(chunk + tid) * 3;
            const float x = p[0], y = p[1], z = p[2];
            sX[tid] = x; sY[tid] = y; sZ[tid] = z;
            sR2[tid] = x * x + y * y + z * z;
        }
        __syncthreads();

        // B (4x16 f32), tile columns n0..n0+15 (mirror of A layout).
        // Branchless: unconditional loads, cndmask selects (no exec splits).
        const int n0 = wv * 16;
        const float bxv = hi ? sZ[n0 + m] : sX[n0 + m];
        const float byv = sY[n0 + m];
        v2f bvv;
        bvv.x = bxv;
        bvv.y = hi ? 0.0f : byv;

        // C = q2[M] + r2[N]
        const float r2n = sR2[n0 + m];
        v8f cc;
#pragma unroll
        for (int r = 0; r < 8; ++r) cc[r] = q2v[r] + r2n;

        // D = q2 + r2 - 2*q.r  (full 16x16 sqdist tile in one instruction)
        v8f dacc = __builtin_amdgcn_wmma_f32_16x16x4_f32(
            false, av, false, bvv, (short)0, cc, false, false);

        // scatter tile to LDS: row = center (M), col = point (N)
#pragma unroll
        for (int r = 0; r < 8; ++r)
            sDist[(rowOff + r) * PADROW + n0 + m] = dacc[r];
        __syncthreads();

        // selection: wave wv streams row wv (its center), 32 cands at a time
#pragma unroll
        for (int c8 = 0; c8 < NC / 32; ++c8) {
            const int col = c8 * 32 + lane;
            float cd = sDist[wv * PADROW + col];
            int   ci = chunk + col;
            // kth-best threshold: scalar broadcast from lane 31
            const float kth = __int_as_float(
                __builtin_amdgcn_readlane(__float_as_int(bd), 31));
            if (__any(cd < kth)) {
                bitonicSort32Asc(cd, ci, lane);
                // reverse candidates (lane 31-i), pairwise-min -> bitonic best
                const float rd = lanex_f<31>(cd);
                const int   ri = lanex_i<31>(ci);
                const bool tk = rd < bd;
                bd = tk ? rd : bd;
                bi = tk ? ri : bi;
                bitonicMerge32Asc(bd, bi, lane);
            }
        }
        __syncthreads();
    }

    // gather neighbors + subtract center; lane l holds l-th nearest
    const float4 cw = sCen[wv];
    const float* p = xyz + bbase + (size_t)bi * 3;
    const float ox = p[0] - cw.x, oy = p[1] - cw.y, oz = p[2] - cw.z;
    float* o = out_neigh + (((size_t)b * NGROUP + g0 + wv) * GSIZE + lane) * 3;
    o[0] = ox; o[1] = oy; o[2] = oz;
}

// ---------------------------------------------------------------------------
extern "C" void kernel_launch(void* const* d_in, const int* in_sizes, int n_in,
                              void* d_out, int out_size, void* d_ws, size_t ws_size,
                              hipStream_t stream) {
    (void)in_sizes; (void)n_in; (void)out_size; (void)d_ws; (void)ws_size;
    const float* xyz = (const float*)d_in[0];
    float* out = (float*)d_out;

    float* out_neigh  = out;                                      // (B,512,32,3)
    float* out_center = out + (size_t)BATCH * NGROUP * GSIZE * 3; // (B,512,3)

    fps_kernel<<<BATCH, 1024, 0, stream>>>(xyz, out_center);
    knn_group_kernel<<<BATCH * (NGROUP / 16), 512, 0, stream>>>(
        xyz, out_center, out_neigh);
}